// GCNet_4990751998611
// MI455X (gfx1250) — compile-verified
//
#include <hip/hip_runtime.h>
#include <hip/hip_bf16.h>

typedef __attribute__((ext_vector_type(2))) float v2f;
typedef __attribute__((ext_vector_type(8))) float v8f;

#define N_NODES 50000
#define N_EDGES 1600000
#define IN_CH   256
#define HID_CH  128
#define OUT_CH  64
#define M_TILES (N_NODES / 16)   // 3125, exact

// ---------------------------------------------------------------- degree

__global__ void deg_init_kernel(float* deg, int n) {
    int i = blockIdx.x * blockDim.x + threadIdx.x;
    if (i < n) deg[i] = 1.0f;             // self-loop contributes 1
}

__global__ void deg_count_kernel(const int* __restrict__ dst, float* deg, int e) {
    int i = blockIdx.x * blockDim.x + threadIdx.x;
    if (i < e) atomicAdd(&deg[dst[i]], 1.0f);
}

__global__ void deg_to_dinv_kernel(float* deg, int n) {
    int i = blockIdx.x * blockDim.x + threadIdx.x;
    if (i < n) deg[i] = rsqrtf(deg[i]);   // deg >= 1 always (self loops)
}

// ---------------------------------------------------------------- WMMA GEMM
// out[M x NCOL] = (RELU ? relu(A) : A)[M x K] * W[K x NCOL]
// One wave computes one 16x16 tile, K-step 4 via V_WMMA_F32_16X16X4_F32.
// 4 waves per block share a K x 16 tile of W staged in LDS.
template <int K, int NCOL, bool RELU>
__global__ __launch_bounds__(128) void gemm_wmma_kernel(
    const float* __restrict__ A, const float* __restrict__ W,
    float* __restrict__ out, int mTiles) {
    __shared__ float bs[K * 16];

    const int tid     = threadIdx.x;
    const int colBase = blockIdx.y * 16;

    // Cooperatively stage W[:, colBase:colBase+16] into LDS (row-major [K][16]).
    for (int i = tid; i < K * 16; i += 128) {
        int kk = i >> 4;
        int nn = i & 15;
        bs[i] = W[kk * NCOL + colBase + nn];
    }
    __syncthreads();

    const int wave  = tid >> 5;
    const int lane  = tid & 31;
    const int hi    = lane >> 4;   // lane half selects K pair
    const int lrow  = lane & 15;   // M (for A) / N (for B,C,D) index
    const int tileM = blockIdx.x * 4 + wave;
    if (tileM >= mTiles) return;

    const float* arow = A + (size_t)(tileM * 16 + lrow) * K + 2 * hi;

    v8f acc = {};
    for (int k = 0; k < K; k += 4) {
        // A 16x4 fp32 layout: lanes 0-15 hold K=k..k+1, lanes 16-31 K=k+2..k+3
        v2f a = *(const v2f*)(arow + k);
        if (RELU) { a.x = fmaxf(a.x, 0.0f); a.y = fmaxf(a.y, 0.0f); }
        // B 4x16 fp32 layout mirrors A with N across lanes, K across VGPRs
        v2f b;
        b.x = bs[(k + 2 * hi + 0) * 16 + lrow];
        b.y = bs[(k + 2 * hi + 1) * 16 + lrow];
        acc = __builtin_amdgcn_wmma_f32_16x16x4_f32(
            false, a, false, b, (short)0, acc, false, false);
    }

    // C/D layout: VGPR r holds C[M = r + 8*hi][N = lrow]
    float* orow = out + (size_t)(tileM * 16 + hi * 8) * NCOL + colBase + lrow;
#pragma unroll
    for (int r = 0; r < 8; ++r) orow[(size_t)r * NCOL] = acc[r];
}

// ---------------------------------------------------------------- aggregation

// out[v][c] = bias[c] + dinv[v]^2 * h[v][c]    (self-loop term + bias)
template <int C>
__global__ void init_self_kernel(const float* __restrict__ dinv,
                                 const float* __restrict__ h,
                                 const float* __restrict__ bias,
                                 float* __restrict__ out, int n) {
    int gid = blockIdx.x * blockDim.x + threadIdx.x;
    if (gid >= n * C) return;
    int v = gid / C;
    int c = gid - v * C;
    float di = dinv[v];
    out[gid] = bias[c] + di * di * h[gid];
}

// One thread per (edge, 4-channel quad): coalesced float4 gather from h[src],
// f32 atomic scatter-add into out[dst] (L2-resident).
template <int C>
__global__ void scatter_edges_kernel(const int* __restrict__ src,
                                     const int* __restrict__ dst,
                                     const float* __restrict__ dinv,
                                     const float* __restrict__ h,
                                     float* __restrict__ out, int e) {
    const int QP = C / 4;  // quads per edge (power of two)
    unsigned gid = blockIdx.x * blockDim.x + threadIdx.x;
    if (gid >= (unsigned)e * QP) return;
    unsigned ei = gid / QP;
    unsigned q  = gid - ei * QP;
    int s = src[ei];
    int d = dst[ei];
    float w = dinv[s] * dinv[d];
    const float4 v = *(const float4*)(h + (size_t)s * C + q * 4);
    float* o = out + (size_t)d * C + q * 4;
    atomicAdd(o + 0, w * v.x);
    atomicAdd(o + 1, w * v.y);
    atomicAdd(o + 2, w * v.z);
    atomicAdd(o + 3, w * v.w);
}

// ---------------------------------------------------------------- log_softmax
// One wave32 per row of 64; 2 channels per lane; shuffle reductions.
__global__ void log_softmax64_kernel(float* __restrict__ out, int n) {
    int row  = (blockIdx.x * blockDim.x + threadIdx.x) >> 5;
    int lane = threadIdx.x & 31;
    if (row >= n) return;
    float* r = out + (size_t)row * 64;
    float2 v = *(float2*)(r + lane * 2);
    float m = fmaxf(v.x, v.y);
#pragma unroll
    for (int off = 16; off >= 1; off >>= 1) m = fmaxf(m, __shfl_xor(m, off, 32));
    float s = __expf(v.x - m) + __expf(v.y - m);
#pragma unroll
    for (int off = 16; off >= 1; off >>= 1) s += __shfl_xor(s, off, 32);
    float lse = m + __logf(s);
    r[lane * 2]     = v.x - lse;
    r[lane * 2 + 1] = v.y - lse;
}

// ---------------------------------------------------------------- launch

extern "C" void kernel_launch(void* const* d_in, const int* in_sizes, int n_in,
                              void* d_out, int out_size, void* d_ws, size_t ws_size,
                              hipStream_t stream) {
    const float* x    = (const float*)d_in[0];            // [N, 256]
    const int*   eidx = (const int*)d_in[1];              // [2, E]
    const float* W1   = (const float*)d_in[2];            // [256, 128]
    const float* b1   = (const float*)d_in[3];            // [128]
    const float* W2   = (const float*)d_in[4];            // [128, 64]
    const float* b2   = (const float*)d_in[5];            // [64]
    float*       outp = (float*)d_out;                    // [N, 64]

    const int* src = eidx;             // edge_index[0]
    const int* dst = eidx + N_EDGES;   // edge_index[1]

    // Workspace layout (floats)
    float* ws   = (float*)d_ws;
    float* dinv = ws;                                   // N
    float* h1   = dinv + N_NODES;                       // N * HID
    float* out1 = h1 + (size_t)N_NODES * HID_CH;        // N * HID
    float* h2   = out1 + (size_t)N_NODES * HID_CH;      // N * OUT

    const int T = 256;
    // --- normalization coefficients
    deg_init_kernel<<<(N_NODES + T - 1) / T, T, 0, stream>>>(dinv, N_NODES);
    deg_count_kernel<<<(N_EDGES + T - 1) / T, T, 0, stream>>>(dst, dinv, N_EDGES);
    deg_to_dinv_kernel<<<(N_NODES + T - 1) / T, T, 0, stream>>>(dinv, N_NODES);

    // --- layer 1: h1 = x @ W1
    dim3 g1((M_TILES + 3) / 4, HID_CH / 16);
    gemm_wmma_kernel<IN_CH, HID_CH, false><<<g1, 128, 0, stream>>>(x, W1, h1, M_TILES);
    // out1 = b1 + dinv^2*h1 + scatter edges
    init_self_kernel<HID_CH><<<((size_t)N_NODES * HID_CH + T - 1) / T, T, 0, stream>>>(
        dinv, h1, b1, out1, N_NODES);
    {
        size_t work = (size_t)N_EDGES * (HID_CH / 4);
        scatter_edges_kernel<HID_CH><<<(work + T - 1) / T, T, 0, stream>>>(
            src, dst, dinv, h1, out1, N_EDGES);
    }

    // --- layer 2: h2 = relu(out1) @ W2
    dim3 g2((M_TILES + 3) / 4, OUT_CH / 16);
    gemm_wmma_kernel<HID_CH, OUT_CH, true><<<g2, 128, 0, stream>>>(out1, W2, h2, M_TILES);
    init_self_kernel<OUT_CH><<<((size_t)N_NODES * OUT_CH + T - 1) / T, T, 0, stream>>>(
        dinv, h2, b2, outp, N_NODES);
    {
        size_t work = (size_t)N_EDGES * (OUT_CH / 4);
        scatter_edges_kernel<OUT_CH><<<(work + T - 1) / T, T, 0, stream>>>(
            src, dst, dinv, h2, outp, N_EDGES);
    }

    // --- log_softmax over 64 channels, one wave per row
    log_softmax64_kernel<<<((size_t)N_NODES * 32 + T - 1) / T, T, 0, stream>>>(
        outp, N_NODES);
}